// update_v_17377437680124
// MI455X (gfx1250) — compile-verified
//
#include <hip/hip_runtime.h>

#define NN 50000
#define NE 800000
#define D 128
#define NSTRIPS (NN / 16)          // 3125, exact
#define WAVES_PER_BLK 4
#define LDS_STRIDE 132             // 128 + 4 pad (even -> 8B-aligned v2f reads)
#define SHIFT_LN2 0.69314718055994530942f

typedef __attribute__((ext_vector_type(2))) float v2f;
typedef __attribute__((ext_vector_type(8))) float v8f;

// ---------------- kernel 1: zero the aggregation buffer ----------------
__global__ void zero_agg_kernel(float4* __restrict__ p, int n4) {
    int i = blockIdx.x * blockDim.x + threadIdx.x;
    if (i < n4) p[i] = make_float4(0.f, 0.f, 0.f, 0.f);
}

// Guaranteed HW fp32 atomic (no-return form -> STOREcnt; drained by the
// implicit wait-idle at S_ENDPGM; cross-kernel order via stream serialization).
__device__ __forceinline__ void hw_atomic_add_f32(float* p, float v) {
    asm volatile("global_atomic_add_f32 %0, %1, off"
                 :
                 : "v"(p), "v"(v)
                 : "memory");
}

// ---------------- kernel 2: scatter-sum edges into nodes ----------------
// one wave (32 lanes) per edge; each lane handles a float4 chunk of the
// 128-float edge feature; native global_atomic_add_f32 into agg.
__global__ void scatter_kernel(const float* __restrict__ e,
                               const long long* __restrict__ edge_index,
                               float* __restrict__ agg) {
    int gid = blockIdx.x * blockDim.x + threadIdx.x;
    int eid = gid >> 5;          // edge id (wave-uniform)
    int c   = gid & 31;          // chunk id (lane)
    // eid is uniform across the wave: readfirstlane lets the compiler use a
    // scalar load (s_load_b64) for the destination index.
    int eid_u = __builtin_amdgcn_readfirstlane(eid);
    long long dst = edge_index[(long long)NE + eid_u];  // edge_index[1][eid]
    const float4 val = ((const float4*)(e + (size_t)eid_u * D))[c];
    float* o = agg + (size_t)dst * D + c * 4;
    hw_atomic_add_f32(o + 0, val.x);
    hw_atomic_add_f32(o + 1, val.y);
    hw_atomic_add_f32(o + 2, val.z);
    hw_atomic_add_f32(o + 3, val.w);
}

// shifted softplus, numerically stable
__device__ __forceinline__ float ssp(float x) {
    return fmaxf(x, 0.f) + log1pf(expf(-fabsf(x))) - SHIFT_LN2;
}

// ---------------- kernel 3: fused  ssp(agg@W1^T+b1)@W2^T + b2 + v -------
// one wave per 16-row strip; full fp32 via V_WMMA_F32_16X16X4_F32.
// A frag (16x4 f32): lanes 0-15: {K0,K1}; lanes 16-31: {K2,K3}; M = lane%16
// B frag (4x16 f32): lanes 0-15: {K0,K1}; lanes 16-31: {K2,K3}; N = lane%16
// C/D (16x16 f32): VGPR r: lanes 0-15 -> M=r, lanes 16-31 -> M=r+8; N=lane%16
__global__ __launch_bounds__(WAVES_PER_BLK * 32)
void gemm_fused_kernel(const float* __restrict__ agg,
                       const float* __restrict__ vin,
                       const float* __restrict__ W1,
                       const float* __restrict__ b1,
                       const float* __restrict__ W2,
                       const float* __restrict__ b2,
                       float* __restrict__ out) {
    __shared__ __align__(16) float lds_h[WAVES_PER_BLK][16 * LDS_STRIDE];

    const int lane  = threadIdx.x & 31;
    const int w     = threadIdx.x >> 5;
    const int strip = blockIdx.x * WAVES_PER_BLK + w;
    if (strip >= NSTRIPS) return;

    const int ln   = lane & 15;   // M (for A) / N (for B,C)
    const int half = lane >> 4;   // k-half selector

    const v8f vzero = {0.f, 0.f, 0.f, 0.f, 0.f, 0.f, 0.f, 0.f};
    v8f c[8];
#pragma unroll
    for (int i = 0; i < 8; ++i) c[i] = vzero;

    // ---- layer 1: h = agg @ W1^T ----
    const float* arow = agg + (size_t)(strip * 16 + ln) * D;
    for (int kk = 0; kk < 32; ++kk) {
        const int k0 = kk * 4 + 2 * half;
        v2f a = *(const v2f*)(arow + k0);
#pragma unroll
        for (int nt = 0; nt < 8; ++nt) {
            v2f b = *(const v2f*)(W1 + (size_t)(nt * 16 + ln) * D + k0);
            c[nt] = __builtin_amdgcn_wmma_f32_16x16x4_f32(
                false, a, false, b, (short)0, c[nt], false, false);
        }
    }

    // ---- bias + shifted softplus, stage h into LDS (A-layout reread) ----
#pragma unroll
    for (int nt = 0; nt < 8; ++nt) {
        const int   col = nt * 16 + ln;
        const float bv  = b1[col];
#pragma unroll
        for (int r = 0; r < 8; ++r) {
            lds_h[w][(r + 8 * half) * LDS_STRIDE + col] = ssp(c[nt][r] + bv);
        }
    }
    // same-wave LDS write->read dependency; compiler inserts s_wait_dscnt

#pragma unroll
    for (int i = 0; i < 8; ++i) c[i] = vzero;

    // ---- layer 2: out = h @ W2^T ----
    const float* hrow = &lds_h[w][ln * LDS_STRIDE];
    for (int kk = 0; kk < 32; ++kk) {
        const int k0 = kk * 4 + 2 * half;
        v2f a = *(const v2f*)(hrow + k0);
#pragma unroll
        for (int nt = 0; nt < 8; ++nt) {
            v2f b = *(const v2f*)(W2 + (size_t)(nt * 16 + ln) * D + k0);
            c[nt] = __builtin_amdgcn_wmma_f32_16x16x4_f32(
                false, a, false, b, (short)0, c[nt], false, false);
        }
    }

    // ---- + b2 + residual v, store ----
#pragma unroll
    for (int nt = 0; nt < 8; ++nt) {
        const int   col = nt * 16 + ln;
        const float bv  = b2[col];
#pragma unroll
        for (int r = 0; r < 8; ++r) {
            const int row = strip * 16 + r + 8 * half;
            const size_t idx = (size_t)row * D + col;
            out[idx] = c[nt][r] + bv + vin[idx];
        }
    }
}

extern "C" void kernel_launch(void* const* d_in, const int* in_sizes, int n_in,
                              void* d_out, int out_size, void* d_ws, size_t ws_size,
                              hipStream_t stream) {
    const float*      v  = (const float*)d_in[0];
    const float*      e  = (const float*)d_in[1];
    const long long*  ei = (const long long*)d_in[2];
    const float*      W1 = (const float*)d_in[3];
    const float*      b1 = (const float*)d_in[4];
    const float*      W2 = (const float*)d_in[5];
    const float*      b2 = (const float*)d_in[6];
    float*            out = (float*)d_out;
    float*            agg = (float*)d_ws;   // NN*D floats = 25.6 MB

    // 1) zero agg  (1.6M float4 -> 6250 blocks)
    zero_agg_kernel<<<(NN * D / 4 + 255) / 256, 256, 0, stream>>>(
        (float4*)agg, NN * D / 4);

    // 2) scatter-sum  (800K edges * 32 lanes = 25.6M threads)
    scatter_kernel<<<(NE * 32) / 256, 256, 0, stream>>>(e, ei, agg);

    // 3) fused MLP + residual  (3125 strips / 4 waves per block)
    gemm_fused_kernel<<<(NSTRIPS + WAVES_PER_BLK - 1) / WAVES_PER_BLK,
                        WAVES_PER_BLK * 32, 0, stream>>>(
        agg, v, W1, b1, W2, b2, out);
}